// QueryDepthPoint_40003325395441
// MI455X (gfx1250) — compile-verified
//
#include <hip/hip_runtime.h>
#include <cstdint>
#include <cstddef>

// Problem constants (match reference: B,N,M = 8,16384,2048; NSAMPLE=64; DIS_Z=0.5)
#define QDP_B     8
#define QDP_N     16384
#define QDP_M     2048
#define QDP_NS    64
#define QDP_DISZ  0.5f
#define QDP_BLOCK 256

// 128-bit payload type for the async b128 builtin (diagnostic showed int __vector(4)*)
typedef int v4i __attribute__((ext_vector_type(4)));
typedef __attribute__((address_space(1))) v4i v4i_g;  // global
typedef __attribute__((address_space(3))) v4i v4i_l;  // LDS

__global__ __launch_bounds__(QDP_BLOCK) void
QueryDepthPoint_40003325395441_kernel(const float* __restrict__ xyz1,
                                      const float* __restrict__ xyz2,
                                      long long* __restrict__ idx_out,
                                      int* __restrict__ cnt_out) {
    // Full z1 row for this batch staged in LDS: 16384 floats = 64 KB
    __shared__ float z1s[QDP_N];

    const int tid = threadIdx.x;
    const int blocksPerBatch = QDP_M / QDP_BLOCK;               // 8
    const int b = blockIdx.x / blocksPerBatch;
    const int m = (blockIdx.x % blocksPerBatch) * QDP_BLOCK + tid;

    // xyz1 layout (B,3,N): z row at b*3*N + 2*N
    const float* z1g = xyz1 + ((size_t)b * 3 + 2) * QDP_N;

    // ---- Stage z1 -> LDS via gfx1250 async global->LDS (ASYNCcnt path) ----
#if defined(__gfx1250__) && __has_builtin(__builtin_amdgcn_global_load_async_to_lds_b128)
    #pragma unroll
    for (int i = 0; i < QDP_N / (QDP_BLOCK * 4); ++i) {         // 16 x b128 per thread
        const int off = (i * QDP_BLOCK + tid) * 4;              // float index
        __builtin_amdgcn_global_load_async_to_lds_b128(
            (v4i_g*)(z1g + off), (v4i_l*)(&z1s[off]),
            /*imm offset=*/0, /*cpol=*/0);
    }
    __builtin_amdgcn_s_wait_asynccnt(0);
#else
    for (int i = tid; i < QDP_N / 4; i += QDP_BLOCK)
        ((float4*)z1s)[i] = ((const float4*)z1g)[i];
#endif
    __syncthreads();

    // ---- Per-query ordered scan with early exit at NSAMPLE matches ----
    // xyz2 layout (B,3,M): z at b*3*M + 2*M + m
    const float zq = xyz2[((size_t)b * 3 + 2) * QDP_M + m];
    long long* __restrict__ row = idx_out + ((size_t)b * QDP_M + m) * QDP_NS;

    int cnt = 0;
    for (int j0 = 0; j0 < QDP_N; j0 += 4) {
        if (cnt >= QDP_NS) break;                  // early exit (~230 iters typical)
        // All lanes read the same LDS address -> broadcast, conflict-free
        const float4 z4 = *(const float4*)(&z1s[j0]);
        const float zz[4] = {z4.x, z4.y, z4.z, z4.w};
        #pragma unroll
        for (int k = 0; k < 4; ++k) {
            if (__builtin_fabsf(zz[k] - zq) < QDP_DISZ) {
                const int j = j0 + k;
                if (cnt == 0) {
                    // ball-query convention: first match pre-fills all slots
                    #pragma unroll 4
                    for (int s = 0; s < QDP_NS; ++s) row[s] = (long long)j;
                } else if (cnt < QDP_NS) {
                    row[cnt] = (long long)j;
                }
                ++cnt;
            }
        }
    }
    if (cnt == 0) {
        // reference yields 0 for rows with no match; d_out is poisoned, so clear
        #pragma unroll 4
        for (int s = 0; s < QDP_NS; ++s) row[s] = 0;
    }
    cnt_out[(size_t)b * QDP_M + m] = (cnt < QDP_NS) ? cnt : QDP_NS;
}

extern "C" void kernel_launch(void* const* d_in, const int* in_sizes, int n_in,
                              void* d_out, int out_size, void* d_ws, size_t ws_size,
                              hipStream_t stream) {
    (void)in_sizes; (void)n_in; (void)out_size; (void)d_ws; (void)ws_size;

    const float* xyz1 = (const float*)d_in[0];   // (B,3,N) f32
    const float* xyz2 = (const float*)d_in[1];   // (B,3,M) f32

    // Outputs concatenated flat in return order: idx (B,M,64) int64, then pts_cnt (B,M) int32
    long long* idx_out = (long long*)d_out;
    int* cnt_out = (int*)(idx_out + (size_t)QDP_B * QDP_M * QDP_NS);

    dim3 grid(QDP_B * (QDP_M / QDP_BLOCK));      // 64 workgroups
    QueryDepthPoint_40003325395441_kernel<<<grid, QDP_BLOCK, 0, stream>>>(
        xyz1, xyz2, idx_out, cnt_out);
}